// OrdinalMixtureGCN_10505490006191
// MI455X (gfx1250) — compile-verified
//
#include <hip/hip_runtime.h>
#include <hip/hip_bf16.h>

typedef __attribute__((ext_vector_type(2))) float v2f;
typedef __attribute__((ext_vector_type(8))) float v8f;

#define D_DIM 128
#define O_DIM 64

// ---------------------------------------------------------------------------
// Zero the output accumulators (harness poisons d_out with 0xAA).
__global__ void zero_kernel(float* __restrict__ out, size_t n) {
    size_t i = (size_t)blockIdx.x * blockDim.x + threadIdx.x;
    size_t stride = (size_t)gridDim.x * blockDim.x;
    for (; i < n; i += stride) out[i] = 0.0f;
}

// ---------------------------------------------------------------------------
// Ordinal cumulative weights: cum[r] = sum_{j<=r} w[j], for both u and v sides.
__global__ void cumsum_kernel(const float* __restrict__ wu,
                              const float* __restrict__ wv,
                              float* __restrict__ cu,
                              float* __restrict__ cv,
                              int R, int DO) {
    int idx = blockIdx.x * blockDim.x + threadIdx.x;
    if (idx >= 2 * DO) return;
    const float* src = (idx < DO) ? wu : wv;
    float*       dst = (idx < DO) ? cu : cv;
    int j = (idx < DO) ? idx : idx - DO;
    float acc = 0.0f;
    for (int r = 0; r < R; ++r) {
        acc += src[(size_t)r * DO + j];
        dst[(size_t)r * DO + j] = acc;
    }
}

// ---------------------------------------------------------------------------
// Dense GEMM tile kernel: T = X[N x 128] @ W[128 x 64] using
// V_WMMA_F32_16X16X4_F32. blockIdx.y picks the u (0) or v (1) side.
// Block = 256 threads = 8 waves; each wave owns one 16x16 output tile.
// Block covers 32 rows x 64 cols. W (32 KB) is staged in LDS.
__global__ void gemm_kernel(const float* __restrict__ xu,
                            const float* __restrict__ xv,
                            const float* __restrict__ Wu,  // cumW for this relation
                            const float* __restrict__ Wv,
                            float* __restrict__ tu,
                            float* __restrict__ tv,
                            int NUr, int NVr) {
    const float* X = blockIdx.y ? xv : xu;
    const float* W = blockIdx.y ? Wv : Wu;
    float*       T = blockIdx.y ? tv : tu;
    const int    N = blockIdx.y ? NVr : NUr;

    __shared__ float Wl[D_DIM * O_DIM];  // 32 KB
    for (int i = threadIdx.x; i < D_DIM * O_DIM; i += 256)
        Wl[i] = W[i];
    __syncthreads();

    const int wave = threadIdx.x >> 5;
    const int lane = threadIdx.x & 31;
    const int rowBase = blockIdx.x * 32 + (wave >> 2) * 16;
    const int nBase   = (wave & 3) * 16;
    if (rowBase >= N) return;

    // A fragment addressing (16x4 f32): lanes 0-15 hold K = {k,k+1},
    // lanes 16-31 hold K = {k+2,k+3}, row M = lane & 15.
    const int m    = lane & 15;
    const int koff = (lane >> 4) * 2;
    int rowA = rowBase + m;
    if (rowA >= N) rowA = N - 1;  // clamp (padding rows recompute row N-1)
    const float* xrow = X + (size_t)rowA * D_DIM + koff;

    // B fragment addressing (4x16 f32): lanes 0-15 -> K = {k,k+1},
    // lanes 16-31 -> K = {k+2,k+3}, column N = nBase + (lane & 15).
    const int n = nBase + (lane & 15);

    v8f acc = {};
#pragma unroll
    for (int k = 0; k < D_DIM; k += 4) {
        v2f a = *(const v2f*)(xrow + k);  // contiguous K pair, 8B aligned
        v2f b;
        b.x = Wl[(k + koff) * O_DIM + n];
        b.y = Wl[(k + koff + 1) * O_DIM + n];
        acc = __builtin_amdgcn_wmma_f32_16x16x4_f32(
            /*neg_a=*/false, a, /*neg_b=*/false, b,
            /*c_mod=*/(short)0, acc, /*reuse_a=*/false, /*reuse_b=*/false);
    }

    // C/D layout: VGPR i -> row (rowBase + i + 8*(lane>=16)), col nBase + (lane&15)
    const int mOff = (lane >> 4) * 8;
    const int nCol = lane & 15;
    float* out = T + (size_t)rowBase * O_DIM + nBase;
#pragma unroll
    for (int i = 0; i < 8; ++i) {
        int row = rowBase + i + mOff;
        if (row < N)
            out[(size_t)(i + mOff) * O_DIM + nCol] = acc[i];
    }
}

// ---------------------------------------------------------------------------
// SpMM scatter for one relation: one wave per edge, lane covers columns
// {2*lane, 2*lane+1}. Edge row/col/val are wave-uniform (scalarized by
// the compiler). fp32 atomics without return -> global_atomic_add_f32.
__global__ void scatter_kernel(const int*   __restrict__ rows,
                               const int*   __restrict__ cols,
                               const float* __restrict__ vals,
                               const float* __restrict__ tmp_u,
                               const float* __restrict__ tmp_v,
                               float* __restrict__ z_u,
                               float* __restrict__ z_v,
                               int E) {
    const int lane   = threadIdx.x & 31;
    const int wave   = (blockIdx.x * blockDim.x + threadIdx.x) >> 5;
    const int nWaves = (gridDim.x * blockDim.x) >> 5;
    const int c2 = lane * 2;

    for (int e = wave; e < E; e += nWaves) {
        const int   ru = rows[e];
        const int   cv = cols[e];
        const float v  = vals[e];

        // z_u[ru,:] += v * tmp_v[cv,:]
        v2f tv = *(const v2f*)(tmp_v + (size_t)cv * O_DIM + c2);
        atomicAdd(&z_u[(size_t)ru * O_DIM + c2],     v * tv.x);
        atomicAdd(&z_u[(size_t)ru * O_DIM + c2 + 1], v * tv.y);

        // z_v[cv,:] += v * tmp_u[ru,:]
        v2f tu = *(const v2f*)(tmp_u + (size_t)ru * O_DIM + c2);
        atomicAdd(&z_v[(size_t)cv * O_DIM + c2],     v * tu.x);
        atomicAdd(&z_v[(size_t)cv * O_DIM + c2 + 1], v * tu.y);
    }
}

// ---------------------------------------------------------------------------
__global__ void relu_kernel(float* __restrict__ out, size_t n) {
    size_t i = (size_t)blockIdx.x * blockDim.x + threadIdx.x;
    size_t stride = (size_t)gridDim.x * blockDim.x;
    for (; i < n; i += stride) {
        float x = out[i];
        out[i] = x > 0.0f ? x : 0.0f;
    }
}

// ---------------------------------------------------------------------------
extern "C" void kernel_launch(void* const* d_in, const int* in_sizes, int n_in,
                              void* d_out, int out_size, void* d_ws, size_t ws_size,
                              hipStream_t stream) {
    const float* x_u       = (const float*)d_in[0];
    const float* x_v       = (const float*)d_in[1];
    const int*   sup_rows  = (const int*)d_in[2];
    const int*   sup_cols  = (const int*)d_in[3];
    const float* sup_vals  = (const float*)d_in[4];
    const float* weights_u = (const float*)d_in[5];
    const float* weights_v = (const float*)d_in[6];
    float* out = (float*)d_out;

    const int R  = 5;
    const int DO = D_DIM * O_DIM;
    const int NU = in_sizes[0] / D_DIM;
    const int NV = in_sizes[1] / D_DIM;
    const int E  = in_sizes[2] / R;

    // Workspace layout: cumWu | cumWv | tmp_u(r) | tmp_v(r)  (~51.5 MB)
    float* cumWu = (float*)d_ws;
    float* cumWv = cumWu + (size_t)R * DO;
    float* tmp_u = cumWv + (size_t)R * DO;
    float* tmp_v = tmp_u + (size_t)NU * O_DIM;

    float* z_u = out;
    float* z_v = out + (size_t)NU * O_DIM;

    const size_t totOut = (size_t)(NU + NV) * O_DIM;
    zero_kernel<<<2048, 256, 0, stream>>>(out, totOut);

    cumsum_kernel<<<(2 * DO + 255) / 256, 256, 0, stream>>>(
        weights_u, weights_v, cumWu, cumWv, R, DO);

    const int nMax = (NU > NV) ? NU : NV;
    dim3 ggrid((nMax + 31) / 32, 2);

    for (int r = 0; r < R; ++r) {
        gemm_kernel<<<ggrid, 256, 0, stream>>>(
            x_u, x_v,
            cumWu + (size_t)r * DO, cumWv + (size_t)r * DO,
            tmp_u, tmp_v, NU, NV);

        scatter_kernel<<<2048, 256, 0, stream>>>(
            sup_rows + (size_t)r * E, sup_cols + (size_t)r * E,
            sup_vals + (size_t)r * E,
            tmp_u, tmp_v, z_u, z_v, E);
    }

    relu_kernel<<<2048, 256, 0, stream>>>(out, totOut);
}